// StochasticRGNLoss_31164282699885
// MI455X (gfx1250) — compile-verified
//
#include <hip/hip_runtime.h>
#include <math.h>

typedef __attribute__((ext_vector_type(2))) float v2f;
typedef __attribute__((ext_vector_type(8))) float v8f;

// Wave32 all-lane sum using the CDNA5 matrix pipe for the first folding step.
// A (16x4) = all 0.5, B VGPR0 = B VGPR1 = x  =>  per ISA B layout,
// B(0,n)=B(1,n)=x[n], B(2,n)=B(3,n)=x[n+16]  =>  D[m][n] = x[n] + x[n+16].
// D VGPR0 in lane l holds x[l%16] + x[(l%16)+16]; 4 xor-shuffles finish 16->1.
// Requires EXEC == all 1s (call only from fully-converged code).
__device__ __forceinline__ float wave_sum32(float x) {
    v2f a; a[0] = 0.5f; a[1] = 0.5f;
    v2f b; b[0] = x;    b[1] = x;
    v8f c = {};
    c = __builtin_amdgcn_wmma_f32_16x16x4_f32(
        /*neg_a=*/false, a, /*neg_b=*/false, b,
        /*c_mod=*/(short)0, c, /*reuse_a=*/false, /*reuse_b=*/false);
    float s = c[0];
    s += __shfl_xor(s, 8, 32);
    s += __shfl_xor(s, 4, 32);
    s += __shfl_xor(s, 2, 32);
    s += __shfl_xor(s, 1, 32);
    return s;
}

__device__ __forceinline__ float block_sum(float acc, float* lds) {
    float ws = wave_sum32(acc);              // all lanes of each wave hold wave total
    const int lane = threadIdx.x & 31;
    const int wid  = threadIdx.x >> 5;
    if (lane == 0) lds[wid] = ws;
    __syncthreads();
    float s = 0.0f;
    if (threadIdx.x == 0) {
        const int nw = (int)(blockDim.x >> 5);
        for (int w = 0; w < nw; ++w) s += lds[w];
    }
    return s;                                // valid in thread 0 only
}

__global__ void __launch_bounds__(256)
rgn_partial_kernel(const float* __restrict__ inp,
                   const float* __restrict__ tgt,
                   const int*   __restrict__ left,
                   const int*   __restrict__ right,
                   float*       __restrict__ partials,
                   int np) {
    __shared__ float lds[8];
    const int tid    = (int)(blockIdx.x * blockDim.x + threadIdx.x);
    const int stride = (int)(gridDim.x * blockDim.x);

    float acc = 0.0f;
    for (int i = tid; i < np; i += stride) {
        const size_t l = (size_t)left[i]  * 9u + 3u;   // CA row offset: idx*9 + 3
        const size_t r = (size_t)right[i] * 9u + 3u;

        // 3-float gathers -> global_load_b96
        const float ax = inp[l + 0] - inp[r + 0];
        const float ay = inp[l + 1] - inp[r + 1];
        const float az = inp[l + 2] - inp[r + 2];
        const float bx = tgt[l + 0] - tgt[r + 0];
        const float by = tgt[l + 1] - tgt[r + 1];
        const float bz = tgt[l + 2] - tgt[r + 2];

        const float din = sqrtf(ax * ax + ay * ay + az * az);
        const float dtg = sqrtf(bx * bx + by * by + bz * bz);
        const float e   = din - dtg;
        acc += e * e;
    }

    // All 256 threads reconverged here -> EXEC all 1s for the WMMA reduce.
    const float s = block_sum(acc, lds);
    if (threadIdx.x == 0) partials[blockIdx.x] = s;
}

__global__ void __launch_bounds__(256)
rgn_final_kernel(const float* __restrict__ partials, int nparts,
                 float* __restrict__ out, float inv_np) {
    __shared__ float lds[8];
    float acc = 0.0f;
    for (int i = (int)threadIdx.x; i < nparts; i += (int)blockDim.x)
        acc += partials[i];
    const float s = block_sum(acc, lds);
    if (threadIdx.x == 0) out[0] = s * inv_np;
}

extern "C" void kernel_launch(void* const* d_in, const int* in_sizes, int n_in,
                              void* d_out, int out_size, void* d_ws, size_t ws_size,
                              hipStream_t stream) {
    const float* inp   = (const float*)d_in[0];   // inputs (N,3,3) f32
    const float* tgt   = (const float*)d_in[1];   // target (N,3,3) f32
    const int*   left  = (const int*)d_in[2];     // (NP,) i32
    const int*   right = (const int*)d_in[3];     // (NP,) i32
    const int    np    = in_sizes[2];

    float* partials = (float*)d_ws;               // 2048 floats = 8 KB scratch

    constexpr int BLOCK = 256;                    // 8 wave32s per block
    constexpr int GRID  = 2048;                   // 8 pairs per thread at NP=4.19M

    rgn_partial_kernel<<<GRID, BLOCK, 0, stream>>>(inp, tgt, left, right, partials, np);
    rgn_final_kernel<<<1, BLOCK, 0, stream>>>(partials, GRID, (float*)d_out,
                                              1.0f / (float)np);
}